// HHNet_74500502717130
// MI455X (gfx1250) — compile-verified
//
#include <hip/hip_runtime.h>
#include <math.h>

typedef float v2f __attribute__((ext_vector_type(2)));
typedef float v8f __attribute__((ext_vector_type(8)));

// ---------------- Hodgkin-Huxley constants ----------------
constexpr float GNA = 40.0f, GK = 35.0f, GLK = 0.3f;
constexpr float ENA = 55.0f, EKR = -77.0f, ELK = -65.0f;
constexpr float GS = 1.0f, VSYN = 0.0f, IAPP = 0.0f;
constexpr float VT = -20.0f, KP = 3.0f;
constexpr float AD = 1.0f, AR = 0.1f;
constexpr float DT = 0.1f;

// Problem shape (fixed by the reference's setup_inputs)
constexpr int BB = 64;         // batch
constexpr int TT = 1000;       // timesteps
constexpr int DD = 784;        // input features
constexpr int N1 = 100;        // layer-1 neurons
constexpr int N2 = 10;         // layer-2 neurons
constexpr int MROWS = BB * TT; // 64000 GEMM rows

// ---------------- HH step helpers ----------------
__device__ __forceinline__ float xfer(float V) {
    // sigmoid((V - VT)/KP)
    return 1.0f / (1.0f + __expf(-(V - VT) * (1.0f / KP)));
}

__device__ __forceinline__ float gate_update(float a, float b, float g) {
    float ab = a + b;
    return (a * DT + (1.0f - 0.5f * DT * ab) * g) / (0.5f * DT * ab + 1.0f);
}

__device__ __forceinline__ float hh_step(float zp, float& V, float& m,
                                         float& gn, float& h, float& y) {
    float m3 = m * m * m;
    float n2 = gn * gn;
    float n4 = n2 * n2;
    float p1 = GNA * m3 * h;
    float p2 = GK * n4;
    float G  = 0.5f * DT * (p1 + p2 + GLK + GS * y);
    float E  = p1 * ENA + p2 * EKR + GLK * ELK + GS * VSYN * y;
    float Vn = (V * (1.0f - G) + DT * (E + IAPP)) / (1.0f + G);

    float aN =  0.02f  * (Vn - 25.0f) / (1.0f - __expf((25.0f - Vn) * (1.0f / 9.0f)));
    float aM =  0.182f * (Vn + 35.0f) / (1.0f - __expf((-Vn - 35.0f) * (1.0f / 9.0f)));
    float aH =  0.25f  * __expf((-Vn - 90.0f) * (1.0f / 12.0f));
    float bN = -0.002f * (Vn - 25.0f) / (1.0f - __expf((Vn - 25.0f) * (1.0f / 9.0f)));
    float bM = -0.124f * (Vn + 35.0f) / (1.0f - __expf((Vn + 35.0f) * (1.0f / 9.0f)));
    float bH =  0.25f  * __expf((Vn + 34.0f) * (1.0f / 12.0f));
    if (Vn == 25.0f)  { aN = 0.18f;  bN = 0.018f; }
    if (Vn == -35.0f) { aM = 1.638f; bM = 1.116f; }

    m  = gate_update(aM, bM, m);
    gn = gate_update(aN, bN, gn);
    h  = gate_update(aH, bH, h);
    float az = AD * zp;
    y = (az * DT + (1.0f - 0.5f * DT * (az + AR)) * y) /
        (0.5f * DT * (az + AR) + 1.0f);
    V = Vn;
    return Vn;
}

// ---------------- WMMA f32 GEMM: C(M,N) = A(M,K) * B(N,K)^T ----------------
// One wave computes a 16 x (16*NT) strip: the A fragment is loaded once per
// K-step and fed into NT back-to-back V_WMMA_F32_16X16X4_F32 ops (NT
// accumulators live in VGPRs). NDIM/KDIM are compile-time so the edge-column
// masking reduces to v_cndmask on only the last N-tile (no exec branching).
template <int NT, int NDIM, int KDIM>
__global__ __launch_bounds__(256) void gemm_wmma_f32_wide(
    const float* __restrict__ A,   // (M, KDIM) row-major
    const float* __restrict__ Bm,  // (NDIM, KDIM) row-major (acts as B^T)
    float* __restrict__ C,         // (M, NDIM) row-major
    int M) {
    const int lane = threadIdx.x & 31;
    const int wave = threadIdx.x >> 5;
    const int tm   = blockIdx.x * (blockDim.x >> 5) + wave;
    if (tm >= (M >> 4)) return;       // whole-wave uniform exit

    const int sub  = lane & 15;       // 0..15
    const int kGrp = (lane >> 4) << 1; // 0 or 2  (A: K sub-offset; B: same)
    const int mRow = tm * 16 + sub;

    const float* aPtr = A + (size_t)mRow * KDIM + kGrp;
    const float* bPtr[NT];
    bool nOK[NT];
    #pragma unroll
    for (int t = 0; t < NT; ++t) {
        int nCol = t * 16 + sub;
        nOK[t]   = (nCol < NDIM);               // compile-time true for inner tiles
        int nCl  = nOK[t] ? nCol : (NDIM - 1);  // clamp -> always-safe load
        bPtr[t]  = Bm + (size_t)nCl * KDIM + kGrp;
    }

    v8f acc[NT];
    #pragma unroll
    for (int t = 0; t < NT; ++t) { v8f z = {}; acc[t] = z; }

    for (int k0 = 0; k0 < KDIM; k0 += 4) {
        v2f a = *(const v2f*)(aPtr + k0);        // 8B-aligned global_load_b64
        #pragma unroll
        for (int t = 0; t < NT; ++t) {
            v2f br = *(const v2f*)(bPtr[t] + k0);
            v2f b;
            b.x = nOK[t] ? br.x : 0.0f;          // cndmask only on edge tile
            b.y = nOK[t] ? br.y : 0.0f;
            acc[t] = __builtin_amdgcn_wmma_f32_16x16x4_f32(
                false, a, false, b, (short)0, acc[t], false, false);
        }
    }

    // C/D layout: VGPR r -> M = tm*16 + r + (lane>=16 ? 8 : 0), N = t*16 + sub
    const int mBase = tm * 16 + ((lane >> 4) << 3);
    #pragma unroll
    for (int t = 0; t < NT; ++t) {
        int nCol = t * 16 + sub;
        if (nCol < NDIM) {
            #pragma unroll
            for (int r = 0; r < 8; ++r)
                C[(size_t)(mBase + r) * NDIM + nCol] = acc[t][r];
        }
    }
}

// ---------------- HH temporal scan ----------------
// Each thread owns NPT independent neuron chains (ILP to hide the sequential
// per-step latency: ~6 v_exp + several rcp in the dependent chain).
// Flat index space is B*N; thread j owns j, j+total, j+2*total, ...
template <int NPT>
__global__ __launch_bounds__(256) void hh_scan(
    const float* __restrict__ z,   // (B, T, N) synaptic drive
    float* __restrict__ out,       // (B, T, N) sigmoid transfer of V
    int T, int N, float y0) {
    const int total = (BB * N) / NPT;
    const int tid = blockIdx.x * blockDim.x + threadIdx.x;
    if (tid >= total) return;

    const float* zB[NPT];
    float* oB[NPT];
    float V[NPT], m[NPT], gn[NPT], h[NPT], y[NPT];
    #pragma unroll
    for (int j = 0; j < NPT; ++j) {
        int idx = tid + j * total;
        int b = idx / N, nn = idx % N;
        size_t base = ((size_t)b * T) * N + nn;
        zB[j] = z + base;
        oB[j] = out + base;
        V[j] = -70.0f; m[j] = 0.0f; gn[j] = 0.0f; h[j] = 1.0f; y[j] = y0;
    }

    const float s0 = xfer(-70.0f);
    #pragma unroll
    for (int j = 0; j < NPT; ++j) oB[j][0] = s0;   // t = 0 trace

    for (int t = 1; t < T; ++t) {
        #pragma unroll
        for (int j = 0; j < NPT; ++j) {
            float zp = zB[j][(size_t)(t - 1) * N];
            float Vn = hh_step(zp, V[j], m[j], gn[j], h[j], y[j]);
            oB[j][(size_t)t * N] = xfer(Vn);
        }
    }
}

// ---------------- launch ----------------
extern "C" void kernel_launch(void* const* d_in, const int* in_sizes, int n_in,
                              void* d_out, int out_size, void* d_ws, size_t ws_size,
                              hipStream_t stream) {
    const float* batch = (const float*)d_in[0];  // (64, 1000, 784)
    const float* W1    = (const float*)d_in[1];  // (100, 784)
    const float* W2    = (const float*)d_in[2];  // (10, 100)
    float* out = (float*)d_out;                  // (64, 1000, 10)

    float* z1 = (float*)d_ws;                    // (64, 1000, 100) = 25.6 MB
    float* T1 = z1 + (size_t)MROWS * N1;         // (64, 1000, 100) = 25.6 MB
    float* z2 = T1 + (size_t)MROWS * N1;         // (64, 1000, 10)  =  2.56 MB

    // GEMM1: z1 = batch @ W1^T  (M=64000, K=784, N=100)
    // 4000 row-strips, each wave does all 7 N-tiles; 8 waves/block -> 500 blocks
    {
        int strips = MROWS / 16;
        gemm_wmma_f32_wide<7, N1, DD>
            <<<(strips + 7) / 8, 256, 0, stream>>>(batch, W1, z1, MROWS);
    }
    // HH layer 1 (y0 = 0), 6400 chains, 4 chains/thread -> 1600 threads
    {
        int total = (BB * N1) / 4;
        hh_scan<4><<<(total + 255) / 256, 256, 0, stream>>>(z1, T1, TT, N1, 0.0f);
    }
    // GEMM2: z2 = T1 @ W2^T  (M=64000, K=100, N=10) -> 4000 strips, 1 N-tile
    {
        int strips = MROWS / 16;
        gemm_wmma_f32_wide<1, N2, N1>
            <<<(strips + 7) / 8, 256, 0, stream>>>(T1, W2, z2, MROWS);
    }
    // HH layer 2 (y0 = 1), 640 chains, 2 chains/thread -> 320 threads
    {
        int total = (BB * N2) / 2;
        hh_scan<2><<<(total + 255) / 256, 256, 0, stream>>>(z2, out, TT, N2, 1.0f);
    }
}